// HyperConvNetwork_8830452760855
// MI455X (gfx1250) — compile-verified
//
#include <hip/hip_runtime.h>
#include <math.h>

// ---- problem constants (match reference) ----
#define NN   100000
#define MM   20000
#define NNZ  1000000
#define DD   128
#define NUM_LAYERS 3
#define NEG_SLOPE 0.2f

typedef __attribute__((ext_vector_type(2))) float v2f;
typedef __attribute__((ext_vector_type(8))) float v8f;

// pointer types for the async-to-LDS builtin (per clang-22 diagnostic:
// param is a vector_size(16) int pointer; src in AS1, dst in LDS AS3)
typedef int v4i_b __attribute__((vector_size(16)));
typedef __attribute__((address_space(1))) v4i_b* as1_v4i;
typedef __attribute__((address_space(3))) v4i_b* as3_v4i;

// order-preserving float <-> uint map (for atomicMax on possibly-negative floats)
__device__ __forceinline__ unsigned ford(float f) {
  unsigned u = __float_as_uint(f);
  return (u & 0x80000000u) ? ~u : (u | 0x80000000u);
}
__device__ __forceinline__ float funord(unsigned u) {
  unsigned v = (u & 0x80000000u) ? (u & 0x7fffffffu) : ~u;
  return __uint_as_float(v);
}

// ---------- generic fills / copies ----------
__global__ void hgc_fill_f32(float* __restrict__ p, float v, size_t n) {
  size_t i = (size_t)blockIdx.x * blockDim.x + threadIdx.x;
  if (i < n) p[i] = v;
}
__global__ void hgc_fill_u32(unsigned* __restrict__ p, unsigned v, size_t n) {
  size_t i = (size_t)blockIdx.x * blockDim.x + threadIdx.x;
  if (i < n) p[i] = v;
}
__global__ void hgc_init_cur_acc(const float* __restrict__ src,
                                 float* __restrict__ cur, float* __restrict__ acc, size_t n) {
  size_t i = (size_t)blockIdx.x * blockDim.x + threadIdx.x;
  if (i < n) { float v = src[i]; cur[i] = v; acc[i] = v; }
}
__global__ void hgc_scale(float* __restrict__ p, float s, size_t n) {
  size_t i = (size_t)blockIdx.x * blockDim.x + threadIdx.x;
  if (i < n) p[i] *= s;
}

// ---------- WMMA fp32 GEMM + fused row-dot ----------
// Y[rows,128] = X[rows,128] @ W[128,128];  s[row] = Y[row,:] . attp[0:128]
// W staged once per workgroup in LDS (64KB), async-to-LDS when available.
// One wave per 16-row tile; A tile in registers; 8 col tiles x 32 k-steps of
// V_WMMA_F32_16X16X4_F32; s computed from accumulators via shfl_xor reduce.
__global__ __launch_bounds__(256) void hgc_gemm128_wmma(
    const float* __restrict__ X, const float* __restrict__ W,
    const float* __restrict__ attp,
    float* __restrict__ Y, float* __restrict__ s, int rows) {
  __shared__ float lw[DD * DD];  // 64 KB

  // ---- stage W into LDS ----
  {
    const float4* src = (const float4*)W;
    float4* dst = (float4*)lw;
#if defined(__gfx1250__) && __has_builtin(__builtin_amdgcn_global_load_async_to_lds_b128)
    for (int i = threadIdx.x; i < DD * DD / 4; i += blockDim.x) {
      __builtin_amdgcn_global_load_async_to_lds_b128(
          (as1_v4i)(v4i_b*)(src + i),
          (as3_v4i)(v4i_b*)(dst + i), 0, 0);
    }
#if __has_builtin(__builtin_amdgcn_s_wait_asynccnt)
    __builtin_amdgcn_s_wait_asynccnt(0);
#else
    asm volatile("s_wait_asynccnt 0" ::: "memory");
#endif
#else
    for (int i = threadIdx.x; i < DD * DD / 4; i += blockDim.x) dst[i] = src[i];
#endif
  }
  __syncthreads();

  int wave = blockIdx.x * (blockDim.x >> 5) + (threadIdx.x >> 5);
  int lane = threadIdx.x & 31;
  int row0 = wave * 16;
  if (row0 >= rows) return;
  int h   = lane >> 4;   // half: 0 or 1
  int l15 = lane & 15;

  // A fragments: vgpr v of step k holds X[row0+l15][4k + 2h + v]
  const float* ap = X + (size_t)(row0 + l15) * DD + 2 * h;
  v2f afrag[32];
#pragma unroll
  for (int k = 0; k < 32; ++k) {
    afrag[k].x = ap[4 * k];
    afrag[k].y = ap[4 * k + 1];
  }

  float sp[8];
#pragma unroll
  for (int v = 0; v < 8; ++v) sp[v] = 0.f;

  for (int c = 0; c < 8; ++c) {
    v8f acc = {};
    // B fragment from LDS: vgpr v of step k holds W[4k + 2h + v][16c + l15]
    const float* bp = lw + (size_t)(2 * h) * DD + c * 16 + l15;
#pragma unroll
    for (int k = 0; k < 32; ++k) {
      v2f b;
      b.x = bp[(4 * k) * DD];
      b.y = bp[(4 * k + 1) * DD];
      acc = __builtin_amdgcn_wmma_f32_16x16x4_f32(
          false, afrag[k], false, b, (short)0, acc, false, false);
    }
    // store D: vgpr v -> (M = v + 8h, N = 16c + l15)
    float* yp = Y + (size_t)(row0 + 8 * h) * DD + c * 16 + l15;
    float av = attp[c * 16 + l15];
#pragma unroll
    for (int v = 0; v < 8; ++v) {
      yp[(size_t)v * DD] = acc[v];
      sp[v] += acc[v] * av;
    }
  }

  // reduce sp over the 16 lanes of each half (masks < 16 stay in-half)
#pragma unroll
  for (int v = 0; v < 8; ++v) {
    float t = sp[v];
    t += __shfl_xor(t, 1, 32);
    t += __shfl_xor(t, 2, 32);
    t += __shfl_xor(t, 4, 32);
    t += __shfl_xor(t, 8, 32);
    if (l15 == 0) s[row0 + 8 * h + v] = t;   // lanes 0/16 -> M = v / v+8
  }
}

// ---------- degree accumulation ----------
__global__ void hgc_degrees(const int* __restrict__ ni, const int* __restrict__ ei,
                            const float* __restrict__ w,
                            float* __restrict__ ddeg, float* __restrict__ bdeg, int nnz) {
  int i = blockIdx.x * blockDim.x + threadIdx.x;
  if (i >= nnz) return;
  atomicAdd(&ddeg[ni[i]], w[ei[i]]);
  atomicAdd(&bdeg[ei[i]], 1.0f);
}
__global__ void hgc_invert(float* __restrict__ v, int n) {
  int i = blockIdx.x * blockDim.x + threadIdx.x;
  if (i >= n) return;
  float x = v[i];
  v[i] = (x > 0.f) ? (1.0f / x) : 0.0f;
}

// ---------- attention: logits + segment max ----------
__global__ void hgc_score_max(const float* __restrict__ s_n, const float* __restrict__ s_e,
                              const int* __restrict__ ni, const int* __restrict__ ei,
                              float* __restrict__ ex, unsigned* __restrict__ segmax, int nnz) {
  int i = blockIdx.x * blockDim.x + threadIdx.x;
  if (i >= nnz) return;
  float s = s_n[ni[i]] + s_e[ei[i]];
  s = (s >= 0.f) ? s : NEG_SLOPE * s;
  ex[i] = s;
  atomicMax(&segmax[ni[i]], ford(s));
}
// ---------- attention: exp + segment sum ----------
__global__ void hgc_exp_sum(const int* __restrict__ ni,
                            const unsigned* __restrict__ segmax,
                            float* __restrict__ ex, float* __restrict__ den, int nnz) {
  int i = blockIdx.x * blockDim.x + threadIdx.x;
  if (i >= nnz) return;
  int n = ni[i];
  float e = expf(ex[i] - funord(segmax[n]));
  ex[i] = e;
  atomicAdd(&den[n], e);
}

// ---------- phase B: edge_out[e,:] += alpha*Binv[e] * xt[n,:] ----------
// One wave owns 32 nnz: coalesced metadata load, shfl broadcast, then per-nnz
// float4 row gather + 4 L2-resident f32 atomics per lane.
__global__ __launch_bounds__(256) void hgc_scatter_edge(
    const float* __restrict__ xt, const float* __restrict__ ex,
    const float* __restrict__ den, const float* __restrict__ binv,
    const int* __restrict__ ni, const int* __restrict__ ei,
    float* __restrict__ edge_out, int nnz) {
  int wave = (int)((blockIdx.x * blockDim.x + threadIdx.x) >> 5);
  int lane = threadIdx.x & 31;
  int base = wave * 32;
  if (base >= nnz) return;
  int idx = base + lane;
  int n = 0, e = 0;
  float coeff = 0.f;
  if (idx < nnz) {
    n = ni[idx];
    e = ei[idx];
    coeff = ex[idx] / (den[n] + 1e-16f) * binv[e];
  }
  int cnt = nnz - base; if (cnt > 32) cnt = 32;
  for (int j = 0; j < cnt; ++j) {
    int   nj = __shfl(n, j, 32);
    int   ej = __shfl(e, j, 32);
    float cj = __shfl(coeff, j, 32);
    float4 v = ((const float4*)(xt + (size_t)nj * DD))[lane];
    float* dst = edge_out + (size_t)ej * DD + lane * 4;
    atomicAdd(dst + 0, cj * v.x);
    atomicAdd(dst + 1, cj * v.y);
    atomicAdd(dst + 2, cj * v.z);
    atomicAdd(dst + 3, cj * v.w);
  }
}

// ---------- phase C: out[n,:] += alpha*Dinv[n] * edge_out[e,:] ----------
__global__ __launch_bounds__(256) void hgc_scatter_node(
    const float* __restrict__ edge_out, const float* __restrict__ ex,
    const float* __restrict__ den, const float* __restrict__ dinv,
    const int* __restrict__ ni, const int* __restrict__ ei,
    float* __restrict__ out, int nnz) {
  int wave = (int)((blockIdx.x * blockDim.x + threadIdx.x) >> 5);
  int lane = threadIdx.x & 31;
  int base = wave * 32;
  if (base >= nnz) return;
  int idx = base + lane;
  int n = 0, e = 0;
  float coeff = 0.f;
  if (idx < nnz) {
    n = ni[idx];
    e = ei[idx];
    coeff = ex[idx] / (den[n] + 1e-16f) * dinv[n];
  }
  int cnt = nnz - base; if (cnt > 32) cnt = 32;
  for (int j = 0; j < cnt; ++j) {
    int   nj = __shfl(n, j, 32);
    int   ej = __shfl(e, j, 32);
    float cj = __shfl(coeff, j, 32);
    float4 v = ((const float4*)(edge_out + (size_t)ej * DD))[lane];
    float* dst = out + (size_t)nj * DD + lane * 4;
    atomicAdd(dst + 0, cj * v.x);
    atomicAdd(dst + 1, cj * v.y);
    atomicAdd(dst + 2, cj * v.z);
    atomicAdd(dst + 3, cj * v.w);
  }
}

// ---------- residual + bias + running stack sum ----------
__global__ void hgc_combine(float* __restrict__ cur, const float* __restrict__ outb,
                            const float* __restrict__ bias, float* __restrict__ acc, size_t n) {
  size_t i = (size_t)blockIdx.x * blockDim.x + threadIdx.x;
  if (i >= n) return;
  float c = cur[i] + outb[i] + bias[i & (DD - 1)];
  cur[i] = c;
  acc[i] += c;
}

static inline int cdiv_i(long long a, long long b) { return (int)((a + b - 1) / b); }

extern "C" void kernel_launch(void* const* d_in, const int* in_sizes, int n_in,
                              void* d_out, int out_size, void* d_ws, size_t ws_size,
                              hipStream_t stream) {
  const float* pois = (const float*)d_in[0];
  const float* traj = (const float*)d_in[1];
  const float* hw   = (const float*)d_in[2];
  const int*   ni   = (const int*)d_in[3];
  const int*   ei   = (const int*)d_in[4];
  const float* Wl   = (const float*)d_in[5];
  const float* att  = (const float*)d_in[6];
  const float* bias = (const float*)d_in[7];
  float* acc = (float*)d_out;   // running stack sum lives in d_out

  const size_t ND = (size_t)NN * DD, MD = (size_t)MM * DD;
  float* ws       = (float*)d_ws;
  float* cur      = ws;                 // ND
  float* xt       = cur + ND;           // ND  (reused as 'out' scatter target)
  float* et       = xt + ND;            // MD
  float* edge_out = et + MD;            // MD
  float* ex       = edge_out + MD;      // NNZ
  float* s_n      = ex + NNZ;           // N
  float* s_e      = s_n + NN;           // M
  unsigned* segmax = (unsigned*)(s_e + MM); // N
  float* den      = (float*)(segmax + NN);  // N
  float* dinv     = den + NN;           // N
  float* binv     = dinv + NN;          // M

  const int B = 256;
  const int gND   = cdiv_i(ND, B);
  const int gMD   = cdiv_i(MD, B);
  const int gNNZ  = cdiv_i(NNZ, B);
  const int gN    = cdiv_i(NN, B);
  const int gM    = cdiv_i(MM, B);
  const int gGemN = cdiv_i((long long)(NN / 16) * 32, B); // one wave per 16-row tile
  const int gGemM = cdiv_i((long long)(MM / 16) * 32, B);
  const int gScat = cdiv_i((long long)cdiv_i(NNZ, 32) * 32, B); // one wave per 32 nnz

  // ---- layer-invariant precompute ----
  hgc_fill_f32<<<gN, B, 0, stream>>>(dinv, 0.f, NN);
  hgc_fill_f32<<<gM, B, 0, stream>>>(binv, 0.f, MM);
  hgc_degrees<<<gNNZ, B, 0, stream>>>(ni, ei, hw, dinv, binv, NNZ);
  hgc_invert<<<gN, B, 0, stream>>>(dinv, NN);
  hgc_invert<<<gM, B, 0, stream>>>(binv, MM);
  hgc_gemm128_wmma<<<gGemM, B, 0, stream>>>(traj, Wl, att + DD, et, s_e, MM);
  hgc_init_cur_acc<<<gND, B, 0, stream>>>(pois, cur, acc, ND);

  // ---- layers ----
  for (int layer = 0; layer < NUM_LAYERS; ++layer) {
    hgc_gemm128_wmma<<<gGemN, B, 0, stream>>>(cur, Wl, att, xt, s_n, NN);

    hgc_fill_u32<<<gN, B, 0, stream>>>(segmax, 0u, NN);   // ford-encoded -inf
    hgc_fill_f32<<<gN, B, 0, stream>>>(den, 0.f, NN);
    hgc_fill_f32<<<gMD, B, 0, stream>>>(edge_out, 0.f, MD);

    hgc_score_max<<<gNNZ, B, 0, stream>>>(s_n, s_e, ni, ei, ex, segmax, NNZ);
    hgc_exp_sum<<<gNNZ, B, 0, stream>>>(ni, segmax, ex, den, NNZ);

    hgc_scatter_edge<<<gScat, B, 0, stream>>>(xt, ex, den, binv, ni, ei, edge_out, NNZ);
    // xt is dead now; reuse its buffer as the node-output scatter target
    hgc_fill_f32<<<gND, B, 0, stream>>>(xt, 0.f, ND);
    hgc_scatter_node<<<gScat, B, 0, stream>>>(edge_out, ex, den, dinv, ni, ei, xt, NNZ);

    hgc_combine<<<gND, B, 0, stream>>>(cur, xt, bias, acc, ND);
  }

  hgc_scale<<<gND, B, 0, stream>>>(acc, 1.0f / (NUM_LAYERS + 1), ND);
}